// SDNConv_3856880632136
// MI455X (gfx1250) — compile-verified
//
#include <hip/hip_runtime.h>
#include <hip/hip_bf16.h>

// SDNConv on MI455X (gfx1250, wave32).
// N=100000 nodes, K=8 factors, DK=16, DIN=128, E=1.6M edges per sign.
// Strategy: matmul-then-scatter (linear ops commute), fp32 WMMA 16x16x4,
// global fp32 atomics for the segment sums, fused tanh+normalize epilogue.
// No workspace needed.

#define N_NODES 100000
#define KFAC    8
#define DKV     16
#define DIN     128        // KFAC * DKV
#define NEDGE   1600000
#define WROWS   80         // 5 * DKV rows per factor in W

typedef __attribute__((ext_vector_type(2))) float v2f;
typedef __attribute__((ext_vector_type(8))) float v8f;

// One 16x16 fp32 GEMM with K=16, built from 4 chained V_WMMA_F32_16X16X4_F32.
// A layout: lane l holds row (l&15), K-pair starting at 4c + 2*(l>>4)  -> v2f
// B layout: lane l holds col (l&15), K rows 4c+2*(l>>4) and +1         -> v2f
// arow: pointer to this lane's A row (16 contiguous floats)
// Wblk: pointer to 16x16 B block, row-major with stride 16
__device__ inline v8f wmma_gemm16(const float* __restrict__ arow,
                                  const float* __restrict__ Wblk,
                                  int l15, int half, v8f acc) {
#pragma unroll
  for (int c = 0; c < 4; ++c) {
    const int kk = 4 * c + 2 * half;
    v2f a = *(const v2f*)(arow + kk);
    v2f b;
    b.x = Wblk[(kk + 0) * DKV + l15];
    b.y = Wblk[(kk + 1) * DKV + l15];
    acc = __builtin_amdgcn_wmma_f32_16x16x4_f32(
        /*neg_a=*/false, a, /*neg_b=*/false, b,
        /*c_mod=*/(short)0, acc, /*reuse_a=*/false, /*reuse_b=*/false);
  }
  return acc;
}

// ---------------------------------------------------------------------------
// Kernel 1: self term.  d_out[n,k,:] = f_in[n,k,:] * W[k, 32:48, :] + b[k,:]
// One wave per (16-node tile, factor k).  Fully initializes d_out.
// ---------------------------------------------------------------------------
__global__ __launch_bounds__(256) void k_self(const float* __restrict__ f_in,
                                              const float* __restrict__ W,
                                              const float* __restrict__ bias,
                                              float* __restrict__ out) {
  const int wave  = (blockIdx.x * blockDim.x + threadIdx.x) >> 5;
  const int lane  = threadIdx.x & 31;
  const int k     = wave & (KFAC - 1);
  const int node0 = (wave >> 3) * 16;      // 6250 tiles exactly
  const int l15   = lane & 15;
  const int half  = lane >> 4;

  const float* arow = f_in + (node0 + l15) * DIN + k * DKV;
  const float* Wblk = W + k * (WROWS * DKV) + 2 * DKV * DKV;  // slot t=2 (self)

  v8f acc = {};
  acc = wmma_gemm16(arow, Wblk, l15, half, acc);

  const float bv = bias[k * DKV + l15];
#pragma unroll
  for (int v = 0; v < 8; ++v) {
    // C/D layout: VGPR v, lane l -> row (v + 8*half), col (l&15)
    out[(node0 + v + 8 * half) * DIN + k * DKV + l15] = acc[v] + bv;
  }
}

// ---------------------------------------------------------------------------
// Kernel 2: edge scatter.  For 16 edges and one factor k, both directions:
//   dir0: out[e.col0] += f_in[e.col1] * W[k, tbase*16   block]
//   dir1: out[e.col1] += f_in[e.col0] * W[k, (tbase+1)  block]
// tbase = 0 for positive edges (p_out/p_in), 3 for negative (n_out/n_in).
// A rows are gathered; D rows are scattered via global fp32 atomics.
// ---------------------------------------------------------------------------
__global__ __launch_bounds__(256) void k_edge(const int* __restrict__ edges,
                                              const float* __restrict__ f_in,
                                              const float* __restrict__ W,
                                              float* __restrict__ out,
                                              int tbase) {
  const int wave = (blockIdx.x * blockDim.x + threadIdx.x) >> 5;
  const int lane = threadIdx.x & 31;
  const int k    = wave & (KFAC - 1);
  const int e0   = (wave >> 3) * 16;       // 100000 groups exactly
  const int l15  = lane & 15;
  const int half = lane >> 4;

  // This lane's edge (row l15 of the A tile): (col0, col1)
  const int ia = edges[2 * (e0 + l15) + 0];
  const int ja = edges[2 * (e0 + l15) + 1];

#pragma unroll
  for (int dir = 0; dir < 2; ++dir) {
    const int src = dir ? ia : ja;         // dir0 gathers col1, dir1 gathers col0
    const float* arow = f_in + src * DIN + k * DKV;
    const float* Wblk = W + k * (WROWS * DKV) + (tbase + dir) * DKV * DKV;

    v8f acc = {};
    acc = wmma_gemm16(arow, Wblk, l15, half, acc);

#pragma unroll
    for (int v = 0; v < 8; ++v) {
      const int m   = v + 8 * half;        // output row within the tile
      const int dst = edges[2 * (e0 + m) + (dir ? 1 : 0)];
      atomicAdd(&out[dst * DIN + k * DKV + l15], acc[v]);
    }
  }
}

// ---------------------------------------------------------------------------
// Kernel 3: epilogue.  a = tanh(f_out); a /= max(||a||_2 over factor axis, eps)
// One thread per (node, dk) pair; reads/writes its own 8 strided floats.
// ---------------------------------------------------------------------------
__global__ __launch_bounds__(256) void k_finish(float* __restrict__ out) {
  const int t = blockIdx.x * blockDim.x + threadIdx.x;
  if (t >= N_NODES * DKV) return;
  const int n = t >> 4;
  const int d = t & 15;
  float* p = out + n * DIN + d;

  float x[KFAC];
  float s = 0.f;
#pragma unroll
  for (int kk = 0; kk < KFAC; ++kk) {
    x[kk] = tanhf(p[kk * DKV]);
    s += x[kk] * x[kk];
  }
  const float inv = 1.0f / fmaxf(sqrtf(s), 1e-12f);
#pragma unroll
  for (int kk = 0; kk < KFAC; ++kk) p[kk * DKV] = x[kk] * inv;
}

extern "C" void kernel_launch(void* const* d_in, const int* in_sizes, int n_in,
                              void* d_out, int out_size, void* d_ws, size_t ws_size,
                              hipStream_t stream) {
  const float* f_in = (const float*)d_in[0];
  const int*   Ep   = (const int*)d_in[1];
  const int*   En   = (const int*)d_in[2];
  const float* W    = (const float*)d_in[3];
  const float* b    = (const float*)d_in[4];
  float* out = (float*)d_out;

  // Kernel 1: 6250 tiles * 8 factors = 50000 waves = 6250 blocks of 8 waves.
  k_self<<<(N_NODES / 16) * KFAC / 8, 256, 0, stream>>>(f_in, W, b, out);

  // Kernel 2: 100000 edge-groups * 8 factors = 800000 waves = 100000 blocks.
  const int egrid = (NEDGE / 16) * KFAC / 8;
  k_edge<<<egrid, 256, 0, stream>>>(Ep, f_in, W, out, /*tbase=*/0);
  k_edge<<<egrid, 256, 0, stream>>>(En, f_in, W, out, /*tbase=*/3);

  // Kernel 3: one thread per (node, dk).
  const int nt = N_NODES * DKV;
  k_finish<<<(nt + 255) / 256, 256, 0, stream>>>(out);
}